// MultiheadAttentionPooling_77807627534808
// MI455X (gfx1250) — compile-verified
//
#include <hip/hip_runtime.h>
#include <math.h>

typedef __attribute__((ext_vector_type(2))) float v2f;
typedef __attribute__((ext_vector_type(8))) float v8f;
typedef __attribute__((ext_vector_type(4))) unsigned int u32x4;
typedef __attribute__((ext_vector_type(8))) int i32x8;
typedef __attribute__((ext_vector_type(4))) int i32x4;

#define B_   32
#define N_   4096
#define DU   64
#define D_   128
#define TILE 128
#define NT   (N_ / TILE)   // 32 tiles per batch
#define PAD_U 68           // padded LDS row (bank-conflict-free 16-row gathers)
#define PAD_D 132
#define PART_OFF 512       // float offset of partials region in ws

// tanh-approx gelu, branch-free: 0.5x(1+tanh(s)) == x * sigmoid(2s).
// Lowers to v_exp_f32 + v_rcp_f32 (TRANS ops, co-executable), no libm slow path.
__device__ __forceinline__ float gelu_tanh(float x) {
    const float c2 = 1.5957691216057308f;      // 2*sqrt(2/pi)
    float g = c2 * (x + 0.044715f * x * x * x);
    return x * __builtin_amdgcn_rcpf(1.0f + __expf(-g));
}

#if __has_builtin(__builtin_amdgcn_tensor_load_to_lds) && __has_builtin(__builtin_amdgcn_s_wait_tensorcnt)
#define USE_TDM 1
#else
#define USE_TDM 0
#endif

#if USE_TDM
// ---- Tensor DMA descriptor packing (CDNA5 D#, 1-D tile, 4-byte elements) ----
__device__ __forceinline__ u32x4 tdm_group0(unsigned lds_addr, const void* gaddr) {
    unsigned long long ga = (unsigned long long)gaddr;
    u32x4 g;
    g.x = 1u;                                            // count=1, user descriptor
    g.y = lds_addr;                                      // LDS byte address
    g.z = (unsigned)(ga & 0xffffffffu);                  // global_addr[31:0]
    g.w = (unsigned)((ga >> 32) & 0x01ffffffu)           // global_addr[56:32]
        | (2u << 30);                                    // type = 2 ("image")
    return g;
}
// n = elements (dwords); optional LDS padding: pad_amount+1 dwords per 2^(iv+1) dwords
__device__ __forceinline__ i32x8 tdm_group1_1d(unsigned n, int pad_en, int iv, int amt) {
    i32x8 g = {};
    g[0] = (2 << 16) | (pad_en << 20) | (iv << 22) | (amt << 25); // data_size=4B
    g[1] = (int)((n & 0xffffu) << 16);                   // tensor_dim0[15:0]
    g[2] = (int)((n >> 16) & 0xffffu) | (1 << 16);       // tensor_dim0[31:16], tensor_dim1=1
    g[3] = (int)((n & 0xffffu) << 16);                   // tile_dim0
    g[4] = 0;                                            // tile_dim1=0, tile_dim2=0 (unused)
    g[5] = (int)n;                                       // tensor_dim0_stride[31:0]
    return g;
}
__device__ __forceinline__ unsigned lds_off(const void* p) {
    // generic LDS pointers carry the LDS byte offset in their low 32 bits
    return (unsigned)(unsigned long long)p;
}
#endif

// ---------------------------------------------------------------------------
// Kernel 1: fold the (batch-independent) pooling query through Wk.
//   q = embed @ Wq + bq; a_h = Wk_h q_h -> ws[0..255]; c_h = bk_h.q_h -> ws[256..257]
// ---------------------------------------------------------------------------
__global__ void prep_kernel(const float* __restrict__ embed,
                            const float* __restrict__ Wq,
                            const float* __restrict__ bq,
                            const float* __restrict__ Wk,
                            const float* __restrict__ bk,
                            float* __restrict__ ws) {
    __shared__ float qv[D_];
    int t = threadIdx.x;                     // 128 threads
    float q = bq[t];
    for (int k = 0; k < D_; ++k) q += embed[k] * Wq[k * D_ + t];
    qv[t] = q;
    __syncthreads();
    for (int h = 0; h < 2; ++h) {
        float a = 0.f;
        for (int e = 0; e < 64; ++e) a += Wk[t * D_ + h * 64 + e] * qv[h * 64 + e];
        ws[h * D_ + t] = a;
    }
    if (t < 2) {
        float cc = 0.f;
        for (int e = 0; e < 64; ++e) cc += bk[t * 64 + e] * qv[t * 64 + e];
        ws[256 + t] = cc;
    }
}

// ---------------------------------------------------------------------------
// Kernel 2: TDM-staged fused MLP (two WMMA f32 GEMMs + gelu) + rank-2 pooling.
// One block = 128 tokens of one batch; 8 waves, each owns a 16-row strip.
// Row-major W1/W2 are DMA-staged into an overlay on s_z1/s_z2, transposed once
// into s_w1t/s_w2t so every B-fragment in the hot loop is a single ds_load_b64
// (the compiler further fuses pairs into ds_load_2addr_b64).
// ---------------------------------------------------------------------------
__global__ void __launch_bounds__(256, 1)
fused_mlp_pool(const float* __restrict__ u,
               const float* __restrict__ W1, const float* __restrict__ b1,
               const float* __restrict__ W2, const float* __restrict__ b2,
               float* __restrict__ ws) {
    __shared__ __align__(16) float s_u  [TILE][PAD_U];   // u tile (padded rows)
    __shared__ __align__(16) float s_w1t[D_ ][PAD_U];    // W1^T [col][k], k<64
    __shared__ __align__(16) float s_w2t[D_ ][PAD_D];    // W2^T [col][k], k<128
    __shared__ __align__(16) float s_z1 [TILE][PAD_D];   // z1; staging overlay for W1
    __shared__ __align__(16) float s_z2 [TILE][PAD_D];   // z2; staging overlay for W2
    __shared__ float s_b1[D_], s_b2[D_];
    __shared__ float s_av[2][D_];
    __shared__ float s_c[2];
    __shared__ float s_wt[2][TILE];

    const int tid  = threadIdx.x;
    const int b    = blockIdx.x >> 5;        // 32 tiles per batch
    const int tile = blockIdx.x & 31;

    float* stW1 = &s_z1[0][0];               // row-major [64][128] staging
    float* stW2 = &s_z2[0][0];               // row-major [128][128] staging
    const float* utile = u + ((size_t)b * N_ + (size_t)tile * TILE) * DU;

#if USE_TDM
    if (tid < 32) {                          // wave 0 drives the Tensor Data Mover
        i32x4 z4 = {};
        i32x8 z8 = {};
        // u tile: 8192 dwords, pad 4 dwords after every 64 (-> [128][68] layout)
        __builtin_amdgcn_tensor_load_to_lds(
            tdm_group0(lds_off(&s_u[0][0]), utile),
            tdm_group1_1d(TILE * DU, 1, 5, 3), z4, z4, z8, 0);
        // W1: 8192 dwords packed row-major
        __builtin_amdgcn_tensor_load_to_lds(
            tdm_group0(lds_off(stW1), W1),
            tdm_group1_1d(DU * D_, 0, 0, 0), z4, z4, z8, 0);
        // W2: 16384 dwords packed row-major
        __builtin_amdgcn_tensor_load_to_lds(
            tdm_group0(lds_off(stW2), W2),
            tdm_group1_1d(D_ * D_, 0, 0, 0), z4, z4, z8, 0);
        __builtin_amdgcn_s_wait_tensorcnt(0);
    }
#else
    {
        const float4* up = (const float4*)utile;
        for (int i = tid; i < TILE * (DU / 4); i += 256) {
            int row = i >> 4, c4 = i & 15;
            *(float4*)&s_u[row][c4 * 4] = up[row * 16 + c4];
        }
        const float4* w1p = (const float4*)W1;
        for (int i = tid; i < DU * (D_ / 4); i += 256)
            *(float4*)&stW1[i * 4] = w1p[i];
        const float4* w2p = (const float4*)W2;
        for (int i = tid; i < D_ * (D_ / 4); i += 256)
            *(float4*)&stW2[i * 4] = w2p[i];
    }
#endif
    if (tid < D_) { s_b1[tid] = b1[tid]; s_b2[tid] = b2[tid]; }
    s_av[tid >> 7][tid & 127] = ws[tid];
    if (tid < 2) s_c[tid] = ws[256 + tid];
    __syncthreads();                         // staging visible to all waves

    // ---- one-time LDS transpose of the weights ----
    for (int i = tid; i < DU * D_; i += 256) {
        int k = i >> 7, c = i & 127;
        s_w1t[c][k] = stW1[k * D_ + c];
    }
    for (int i = tid; i < D_ * D_; i += 256) {
        int k = i >> 7, c = i & 127;
        s_w2t[c][k] = stW2[k * D_ + c];
    }
    __syncthreads();                         // staging fully consumed; overlays reusable

    // ---- WMMA fragment lane mapping (f32 16x16x4) ----
    const int lane   = tid & 31;
    const int wid    = tid >> 5;             // 0..7
    const int rbase  = wid * 16;
    const int arow   = lane & 15;
    const int koff   = (lane >> 4) << 1;     // lanes 16-31 hold K+2,K+3
    const int ncol   = lane & 15;
    const int rowoff = (lane >> 4) << 3;

    // ---- GEMM1: z1 = gelu(u @ W1 + b1), K = 64 ----
#pragma unroll 1
    for (int c = 0; c < D_ / 16; ++c) {
        v8f acc = {};
        const int bc = c * 16 + ncol;
#pragma unroll
        for (int k0 = 0; k0 < DU; k0 += 4) {
            v2f a = *(const v2f*)&s_u  [rbase + arow][k0 + koff];
            v2f w = *(const v2f*)&s_w1t[bc         ][k0 + koff];
            acc = __builtin_amdgcn_wmma_f32_16x16x4_f32(false, a, false, w,
                                                        (short)0, acc, false, false);
        }
        float bias = s_b1[bc];
#pragma unroll
        for (int r = 0; r < 8; ++r)
            s_z1[rbase + rowoff + r][bc] = gelu_tanh(acc[r] + bias);
    }

    // GEMM2 A-rows are the rows this wave just wrote; LDS ops within a wave are
    // in-order (DScnt), so no workgroup barrier is needed between the GEMMs.

    // ---- GEMM2: z2 = gelu(z1 @ W2 + b2), K = 128 ----
#pragma unroll 1
    for (int c = 0; c < D_ / 16; ++c) {
        v8f acc = {};
        const int bc = c * 16 + ncol;
#pragma unroll
        for (int k0 = 0; k0 < D_; k0 += 4) {
            v2f a = *(const v2f*)&s_z1 [rbase + arow][k0 + koff];
            v2f w = *(const v2f*)&s_w2t[bc         ][k0 + koff];
            acc = __builtin_amdgcn_wmma_f32_16x16x4_f32(false, a, false, w,
                                                        (short)0, acc, false, false);
        }
        float bias = s_b2[bc];
#pragma unroll
        for (int r = 0; r < 8; ++r)
            s_z2[rbase + rowoff + r][bc] = gelu_tanh(acc[r] + bias);
    }
    __syncthreads();

    // ---- Galerkin scalar weights: w_h[n] = z2[n] . a_h + c_h ----
    if (tid < TILE) {
        float w0 = s_c[0], w1 = s_c[1];
        for (int d = 0; d < D_; ++d) {
            float zv = s_z2[tid][d];
            w0 += zv * s_av[0][d];
            w1 += zv * s_av[1][d];
        }
        s_wt[0][tid] = w0;
        s_wt[1][tid] = w1;
    }
    __syncthreads();

    // ---- tile partials: S_h[d] = sum_n w_h[n] z2[n][d];  T_h = sum_n w_h[n] ----
    {
        int h = tid >> 7, d = tid & 127;
        float S = 0.f;
        for (int n = 0; n < TILE; ++n) S += s_wt[h][n] * s_z2[n][d];
        float* part = ws + PART_OFF + ((size_t)((b * NT + tile) * 2 + h)) * 129;
        part[d] = S;
        if (d == 0) {
            float ts = 0.f;
            for (int n = 0; n < TILE; ++n) ts += s_wt[h][n];
            part[128] = ts;
        }
    }
}

// ---------------------------------------------------------------------------
// Kernel 3: per-batch deterministic reduction + tiny output projections.
//   out[b] = ((S_h @ Wv_h + T_h * bv_h)/N  concat over h) @ Wo + bo
// ---------------------------------------------------------------------------
__global__ void finish_kernel(const float* __restrict__ Wv, const float* __restrict__ bv,
                              const float* __restrict__ Wo, const float* __restrict__ bo,
                              const float* __restrict__ ws, float* __restrict__ out) {
    __shared__ float Sred[2][D_];
    __shared__ float tred[2];
    __shared__ float attn[D_];
    int b = blockIdx.x, t = threadIdx.x;     // 128 threads
    const float* part = ws + PART_OFF + (size_t)b * NT * 2 * 129;
    for (int h = 0; h < 2; ++h) {
        float s = 0.f;
        for (int tl = 0; tl < NT; ++tl) s += part[(tl * 2 + h) * 129 + t];
        Sred[h][t] = s;
    }
    if (t < 2) {
        float s = 0.f;
        for (int tl = 0; tl < NT; ++tl) s += part[(tl * 2 + t) * 129 + 128];
        tred[t] = s;
    }
    __syncthreads();
    {
        int h = t >> 6;                      // t == h*64 + e
        float a = tred[h] * bv[t];
        for (int d = 0; d < D_; ++d) a += Sred[h][d] * Wv[d * D_ + t];
        attn[t] = a * (1.0f / (float)N_);
    }
    __syncthreads();
    float o = bo[t];
    for (int d = 0; d < D_; ++d) o += attn[d] * Wo[d * D_ + t];
    out[b * D_ + t] = o;
}

extern "C" void kernel_launch(void* const* d_in, const int* in_sizes, int n_in,
                              void* d_out, int out_size, void* d_ws, size_t ws_size,
                              hipStream_t stream) {
    const float* u     = (const float*)d_in[0];
    // d_in[1] = x (unused by the reference)
    const float* W1    = (const float*)d_in[2];
    const float* b1    = (const float*)d_in[3];
    const float* W2    = (const float*)d_in[4];
    const float* b2    = (const float*)d_in[5];
    const float* embed = (const float*)d_in[6];
    const float* Wq    = (const float*)d_in[7];
    const float* bq    = (const float*)d_in[8];
    const float* Wk    = (const float*)d_in[9];
    const float* bk    = (const float*)d_in[10];
    const float* Wv    = (const float*)d_in[11];
    const float* bv    = (const float*)d_in[12];
    const float* Wo    = (const float*)d_in[13];
    const float* bo    = (const float*)d_in[14];
    float* out = (float*)d_out;
    float* ws  = (float*)d_ws;

    prep_kernel<<<1, 128, 0, stream>>>(embed, Wq, bq, Wk, bk, ws);
    fused_mlp_pool<<<B_ * NT, 256, 0, stream>>>(u, W1, b1, W2, b2, ws);
    finish_kernel<<<B_, 128, 0, stream>>>(Wv, bv, Wo, bo, ws, out);
}